// Coulomb_47991964566166
// MI455X (gfx1250) — compile-verified
//
#include <hip/hip_runtime.h>

typedef float v2f __attribute__((ext_vector_type(2)));
typedef float v8f __attribute__((ext_vector_type(8)));

#define COULOMB_CONST 7.1998226

// Each wave owns one 16-row tile of one batch's NxN matrix and sweeps the full
// K = N columns, accumulating D = A x B + C with V_WMMA_F32_16X16X4_F32:
//   A[m,k] = rcp(d[i0+m, k0+k])   (16x4 f32 tile)
//   B[k,n] = q[k0+k]  for all n   (4x16 f32, columns identical)
// => D[m,n] = sum_k rcp(d[m,k]) * q[k]  (identical across n)
// Then fold q_i per row, wave-reduce, and emit one double partial per wave.
__global__ void coulomb_wmma_kernel(const float* __restrict__ d_ij,
                                    const float* __restrict__ q,
                                    double* __restrict__ ws,
                                    int N, int tilesPerBatch, int totalWaves)
{
    const int tid   = blockIdx.x * blockDim.x + threadIdx.x;
    const int wave  = tid >> 5;
    if (wave >= totalWaves) return;            // whole-wave uniform exit

    const int lane   = threadIdx.x & 31;
    const int laneLo = lane & 15;              // M within tile (A/B/D row/col)
    const int laneHi = lane >> 4;              // 0: K=0,1 half   1: K=2,3 half
    const int kOff   = laneHi * 2;

    const int batch = wave / tilesPerBatch;
    const int tile  = wave % tilesPerBatch;
    const int i0    = tile * 16;
    const int row   = i0 + laneLo;

    const float* __restrict__ arow = d_ij + ((size_t)batch * N + row) * (size_t)N + kOff;
    const float* __restrict__ qrow = q + (size_t)batch * N + kOff;

    v8f c = {};
    const int kIters = N >> 4;                 // 16 K-values (4 WMMAs) per iter
    for (int it = 0; it < kIters; ++it) {
        const float* ap = arow + (size_t)it * 16;
        const float* qp = qrow + (size_t)it * 16;

        // 8 independent b64 loads in flight per iteration
        v2f d0 = *(const v2f*)(ap + 0);
        v2f d1 = *(const v2f*)(ap + 4);
        v2f d2 = *(const v2f*)(ap + 8);
        v2f d3 = *(const v2f*)(ap + 12);
        v2f b0 = *(const v2f*)(qp + 0);
        v2f b1 = *(const v2f*)(qp + 4);
        v2f b2 = *(const v2f*)(qp + 8);
        v2f b3 = *(const v2f*)(qp + 12);

        v2f a0, a1, a2, a3;
        a0.x = __builtin_amdgcn_rcpf(d0.x); a0.y = __builtin_amdgcn_rcpf(d0.y);
        a1.x = __builtin_amdgcn_rcpf(d1.x); a1.y = __builtin_amdgcn_rcpf(d1.y);
        a2.x = __builtin_amdgcn_rcpf(d2.x); a2.y = __builtin_amdgcn_rcpf(d2.y);
        a3.x = __builtin_amdgcn_rcpf(d3.x); a3.y = __builtin_amdgcn_rcpf(d3.y);

        // 8 args: (neg_a, A, neg_b, B, c_mod, C, reuse_a, reuse_b)
        c = __builtin_amdgcn_wmma_f32_16x16x4_f32(false, a0, false, b0, (short)0, c, false, false);
        c = __builtin_amdgcn_wmma_f32_16x16x4_f32(false, a1, false, b1, (short)0, c, false, false);
        c = __builtin_amdgcn_wmma_f32_16x16x4_f32(false, a2, false, b2, (short)0, c, false, false);
        c = __builtin_amdgcn_wmma_f32_16x16x4_f32(false, a3, false, b3, (short)0, c, false, false);
    }

    // D layout: VGPR v, lane L -> M = v + (L>=16 ? 8 : 0), N = L&15.
    // Fold q_i per row; every row appears in 16 lanes (one per column N).
    const float* __restrict__ qi = q + (size_t)batch * N + i0 + laneHi * 8;
    float partial = 0.0f;
#pragma unroll
    for (int v = 0; v < 8; ++v)
        partial += qi[v] * c[v];

    // wave32 butterfly reduction
#pragma unroll
    for (int off = 16; off > 0; off >>= 1)
        partial += __shfl_xor(partial, off, 32);

    if (lane == 0)
        ws[wave] = (double)partial * (1.0 / 16.0);   // undo 16x column replication
}

// Deterministic finalize: fixed-order double sum of the per-wave partials.
__global__ void coulomb_finalize(const double* __restrict__ ws,
                                 float* __restrict__ out,
                                 int B, int tilesPerBatch)
{
    int b = blockIdx.x * blockDim.x + threadIdx.x;
    if (b < B) {
        double s = 0.0;
        for (int t = 0; t < tilesPerBatch; ++t)
            s += ws[(size_t)b * tilesPerBatch + t];
        out[b] = (float)(s * (double)COULOMB_CONST);
    }
}

extern "C" void kernel_launch(void* const* d_in, const int* in_sizes, int n_in,
                              void* d_out, int out_size, void* d_ws, size_t ws_size,
                              hipStream_t stream)
{
    const float* d_dij = (const float*)d_in[0];   // [B, N, N] float32
    const float* d_q   = (const float*)d_in[1];   // [B, N]    float32
    float* out = (float*)d_out;                   // [B]

    const long long s0 = in_sizes[0];             // B*N*N
    const long long s1 = in_sizes[1];             // B*N
    const int N = (int)(s0 / s1);                 // 2048
    const int B = (int)(s1 / N);                  // 16

    const int tilesPerBatch = N / 16;             // 128
    const int totalWaves    = B * tilesPerBatch;  // 2048 -> 16 KB of ws doubles

    double* ws = (double*)d_ws;

    const int threads = 256;                                  // 8 waves / block
    const int blocks  = (totalWaves * 32 + threads - 1) / threads;

    hipLaunchKernelGGL(coulomb_wmma_kernel, dim3(blocks), dim3(threads), 0, stream,
                       d_dij, d_q, ws, N, tilesPerBatch, totalWaves);

    hipLaunchKernelGGL(coulomb_finalize, dim3((B + 255) / 256), dim3(256), 0, stream,
                       ws, out, B, tilesPerBatch);
}